// ProjmapSPD_56126632624174
// MI455X (gfx1250) — compile-verified
//
#include <hip/hip_runtime.h>
#include <math.h>

// Matrix log of batched 64x64 SPD matrices via inverse scaling-and-squaring
// with inverse-free Newton-Schulz square roots. All heavy math is f32 WMMA.
//
// Structure exploited: every matrix in the iteration (Y, Z, T, E, Horner S)
// is a polynomial in the same symmetric B0 => symmetric. Hence
//  (a) B-fragments load row-major (contiguous b64, identical to A-fragments),
//  (b) only upper-triangle tiles are computed, mirror-stored to both halves,
//  (c) the it==0 step of each sqrt level (Z==I) needs no Z*Y / T*Z GEMMs.

#define LDA 68   // padded LDS row stride (floats); stride 64 would be 1-bank

typedef __attribute__((ext_vector_type(2))) float v2f;
typedef __attribute__((ext_vector_type(8))) float v8f;

// 16x16 tile (mb,nb) of C = A * B for SYMMETRIC B, A/B in LDS (stride LDA).
// A frag (ISA 32-bit A 16x4): lane<16 -> K=k0,k0+1 ; lane>=16 -> K=k0+2,k0+3.
// B frag: need B[kk][nb+l16]; symmetry => row-major load B[(nb+l16)][kk..kk+1].
// Two independent accumulator chains (even/odd K-groups) for WMMA latency
// hiding; folded with 8 adds at the end.
__device__ __forceinline__ v8f tile_mm_sym(const float* __restrict__ A,
                                           const float* __restrict__ B,
                                           int mb, int nb, int half, int l16) {
  v8f acc0 = {0.f, 0.f, 0.f, 0.f, 0.f, 0.f, 0.f, 0.f};
  v8f acc1 = {0.f, 0.f, 0.f, 0.f, 0.f, 0.f, 0.f, 0.f};
  const float* ap = A + (mb + l16) * LDA + 2 * half;
  const float* bp = B + (nb + l16) * LDA + 2 * half;
#pragma unroll
  for (int k0 = 0; k0 < 64; k0 += 8) {
    const v2f a0 = *(const v2f*)(ap + k0);
    const v2f b0 = *(const v2f*)(bp + k0);
    acc0 = __builtin_amdgcn_wmma_f32_16x16x4_f32(false, a0, false, b0, (short)0,
                                                 acc0, false, false);
    const v2f a1 = *(const v2f*)(ap + k0 + 4);
    const v2f b1 = *(const v2f*)(bp + k0 + 4);
    acc1 = __builtin_amdgcn_wmma_f32_16x16x4_f32(false, a1, false, b1, (short)0,
                                                 acc1, false, false);
  }
#pragma unroll
  for (int r = 0; r < 8; ++r) acc0[r] += acc1[r];
  return acc0;
}

// Store tile and (for off-diagonal tiles) its transpose mirror.
// C/D layout: vgpr r, lane -> row = r + 8*(lane>=16), col = lane%16.
// Mirror addresses are consecutive in r => vectorizable LDS stores.
__device__ __forceinline__ void tile_st_sym(float* __restrict__ D, v8f acc,
                                            int mb, int nb, int half, int l16,
                                            float scale, float diagAdd) {
  const bool off = (mb != nb);
#pragma unroll
  for (int r = 0; r < 8; ++r) {
    const int row = mb + r + 8 * half;
    const int col = nb + l16;
    const float v = scale * acc[r] + ((row == col) ? diagAdd : 0.f);
    D[row * LDA + col] = v;
    if (off) D[col * LDA + row] = v;
  }
}

__global__ __launch_bounds__(128) void logm_spd_wmma_kernel(
    const float* __restrict__ x, float* __restrict__ out, int nmat) {
  __shared__ float Y[64 * LDA];   // current iterate / sqrt chain
  __shared__ float Z[64 * LDA];   // NS dual iterate; later E = R - I
  __shared__ float W[64 * LDA];   // T scratch; later Horner accumulator
  __shared__ float red[64];
  __shared__ float sc[2];

  const int b = blockIdx.x;
  if (b >= nmat) return;  // uniform across block
  const float* __restrict__ xb = x + (size_t)b * 4096;
  float* __restrict__ ob = out + (size_t)b * 4096;

  const int t = threadIdx.x;
  const int lane = t & 31;
  const int half = lane >> 4;
  const int l16 = lane & 15;
  // Provably-scalar wave id so per-wave tile counts become s_cbranch
  const int wv = __builtin_amdgcn_readfirstlane((int)(threadIdx.x >> 5));

  // Upper-triangle tile lists per wave: 10 tiles, counts {3,3,2,2}
  //  w0: (0,0)(0,1)(0,2)  w1: (0,3)(1,1)(1,2)  w2: (1,3)(2,2)  w3: (2,3)(3,3)
  const int ti0[4] = {0, 0, 1, 2}, tj0[4] = {0, 3, 3, 3};
  const int ti1[4] = {0, 1, 2, 3}, tj1[4] = {1, 1, 2, 3};
  const int ti2[4] = {0, 1, 0, 0}, tj2[4] = {2, 2, 0, 0};
  const int TI[3] = {ti0[wv] * 16, ti1[wv] * 16, ti2[wv] * 16};
  const int TJ[3] = {tj0[wv] * 16, tj1[wv] * 16, tj2[wv] * 16};
  const int nt = (wv < 2) ? 3 : 2;

  // ---- symmetrize: Y = 0.5*(X + X^T) ----
  for (int idx = t; idx < 4096; idx += 128) {
    const int i = idx >> 6, j = idx & 63;
    Y[i * LDA + j] = 0.5f * (xb[i * 64 + j] + xb[j * 64 + i]);
  }
  __syncthreads();

  // ---- Gershgorin bound c = max_i sum_j |Y[i][j]| >= lambda_max ----
  if (t < 64) {
    float s = 0.f;
    for (int j = 0; j < 64; ++j) s += fabsf(Y[t * LDA + j]);
    red[t] = s;
  }
  __syncthreads();
  if (t == 0) {
    float m = red[0];
    for (int i = 1; i < 64; ++i) m = fmaxf(m, red[i]);
    m = fmaxf(m, 1e-20f);
    sc[0] = 1.0f / m;
    sc[1] = logf(m);
  }
  __syncthreads();
  const float inv_c = sc[0];
  const float logc = sc[1];

  for (int idx = t; idx < 4096; idx += 128) {
    const int i = idx >> 6, j = idx & 63;
    Y[i * LDA + j] *= inv_c;  // spec(Y) in (0, 1]
  }
  __syncthreads();

  // ---- 7 sqrt levels, coupled Newton-Schulz (inverse-free) ----
  const int iters[7] = {15, 12, 9, 7, 5, 4, 4};
#pragma unroll 1
  for (int lev = 0; lev < 7; ++lev) {
    // it == 0 with Z == I:  T = 1.5I - 0.5Y (elementwise),
    //                       Ynew = Y*T (GEMM), Znew = T (copy).
    for (int idx = t; idx < 4096; idx += 128) {
      const int i = idx >> 6, j = idx & 63;
      W[i * LDA + j] = -0.5f * Y[i * LDA + j] + ((i == j) ? 1.5f : 0.f);
    }
    __syncthreads();
    {
      v8f aY[3];
#pragma unroll
      for (int q = 0; q < 3; ++q)
        if (q < nt) aY[q] = tile_mm_sym(Y, W, TI[q], TJ[q], half, l16);
      // Z = T (reads W, writes Z: no conflict with the GEMM reads above)
      for (int idx = t; idx < 4096; idx += 128) {
        const int i = idx >> 6, j = idx & 63;
        Z[i * LDA + j] = W[i * LDA + j];
      }
      __syncthreads();
#pragma unroll
      for (int q = 0; q < 3; ++q)
        if (q < nt) tile_st_sym(Y, aY[q], TI[q], TJ[q], half, l16, 1.f, 0.f);
      __syncthreads();
    }
#pragma unroll 1
    for (int it = 1; it < iters[lev]; ++it) {
      // T = 1.5 I - 0.5 * Z*Y   -> W  (W not read this phase: store directly)
      v8f aT[3];
#pragma unroll
      for (int q = 0; q < 3; ++q)
        if (q < nt) aT[q] = tile_mm_sym(Z, Y, TI[q], TJ[q], half, l16);
#pragma unroll
      for (int q = 0; q < 3; ++q)
        if (q < nt) tile_st_sym(W, aT[q], TI[q], TJ[q], half, l16, -0.5f, 1.5f);
      __syncthreads();

      // Ynew = Y*T, Znew = T*Z (regs first; Y/Z still being read elsewhere)
      v8f aY[3], aZ[3];
#pragma unroll
      for (int q = 0; q < 3; ++q)
        if (q < nt) {
          aY[q] = tile_mm_sym(Y, W, TI[q], TJ[q], half, l16);
          aZ[q] = tile_mm_sym(W, Z, TI[q], TJ[q], half, l16);
        }
      __syncthreads();
#pragma unroll
      for (int q = 0; q < 3; ++q)
        if (q < nt) {
          tile_st_sym(Y, aY[q], TI[q], TJ[q], half, l16, 1.f, 0.f);
          tile_st_sym(Z, aZ[q], TI[q], TJ[q], half, l16, 1.f, 0.f);
        }
      __syncthreads();
    }
  }

  // ---- E = Y - I (small spec); S0 = 0.2*E - 0.25*I ----
  for (int idx = t; idx < 4096; idx += 128) {
    const int i = idx >> 6, j = idx & 63;
    const float e = Y[i * LDA + j] - ((i == j) ? 1.f : 0.f);
    Z[i * LDA + j] = e;
    W[i * LDA + j] = 0.2f * e + ((i == j) ? -0.25f : 0.f);
  }
  __syncthreads();

  // ---- Horner: S <- c_k I + E*S  for c_k = 1/3, -1/2, 1 ----
  const float hc[3] = {1.f / 3.f, -0.5f, 1.f};
#pragma unroll 1
  for (int s = 0; s < 3; ++s) {
    v8f aP[3];
#pragma unroll
    for (int q = 0; q < 3; ++q)
      if (q < nt) aP[q] = tile_mm_sym(Z, W, TI[q], TJ[q], half, l16);
    __syncthreads();
#pragma unroll
    for (int q = 0; q < 3; ++q)
      if (q < nt) tile_st_sym(W, aP[q], TI[q], TJ[q], half, l16, 1.f, hc[s]);
    __syncthreads();
  }

  // ---- out = 2^7 * E*S + log(c)*I  (mirrored store to global) ----
#pragma unroll
  for (int q = 0; q < 3; ++q)
    if (q < nt) {
      const v8f acc = tile_mm_sym(Z, W, TI[q], TJ[q], half, l16);
      const int mb = TI[q], nb = TJ[q];
#pragma unroll
      for (int r = 0; r < 8; ++r) {
        const int row = mb + r + 8 * half;
        const int col = nb + l16;
        const float v = 128.f * acc[r] + ((row == col) ? logc : 0.f);
        ob[row * 64 + col] = v;
        if (mb != nb) ob[col * 64 + row] = v;
      }
    }
}

extern "C" void kernel_launch(void* const* d_in, const int* in_sizes, int n_in,
                              void* d_out, int out_size, void* d_ws,
                              size_t ws_size, hipStream_t stream) {
  (void)n_in;
  (void)d_ws;
  (void)ws_size;
  (void)out_size;
  const float* x = (const float*)d_in[0];
  float* out = (float*)d_out;
  const int nmat = in_sizes[0] / 4096;  // B = 16384 matrices of 64x64
  logm_spd_wmma_kernel<<<dim3(nmat), dim3(128), 0, stream>>>(x, out, nmat);
}